// GLMTopNRouter_37503654428780
// MI455X (gfx1250) — compile-verified
//
#include <hip/hip_runtime.h>
#include <stdint.h>

typedef __attribute__((ext_vector_type(2))) float v2f;
typedef __attribute__((ext_vector_type(4))) float v4f;
typedef __attribute__((ext_vector_type(8))) float v8f;
typedef __attribute__((ext_vector_type(4))) int   v4i;

constexpr int T_TOK   = 16384;  // tokens
constexpr int H_DIM   = 4096;   // hidden
constexpr int E_EXP   = 64;     // experts
constexpr int BM      = 64;     // tokens per block
constexpr int KT      = 32;     // K tile width (floats)
constexpr int KTP     = 36;     // padded LDS row stride (floats)
constexpr int THREADS = 128;    // 4 waves of 32
constexpr int NT      = H_DIM / KT;  // 128 K-tiles

#if __has_builtin(__builtin_amdgcn_global_load_async_to_lds_b128)
#define ASYNC_LDS 1
#else
#define ASYNC_LDS 0
#endif

#if __has_builtin(__builtin_amdgcn_s_wait_asynccnt)
#define WAIT_ASYNC(n) __builtin_amdgcn_s_wait_asynccnt(n)
#else
#define WAIT_ASYNC(n) asm volatile("s_wait_asynccnt %0" ::"i"(n) : "memory")
#endif

// Address-space casts for the async-copy builtin (prototype: v4i AS1*, v4i AS3*).
#define GPTR(p) ((__attribute__((address_space(1))) v4i*)(uintptr_t)(p))
#define LPTR(p) ((__attribute__((address_space(3))) v4i*)(uint32_t)(uintptr_t)(p))

// ---------------------------------------------------------------------------
// 8 K-steps of V_WMMA_F32_16X16X4_F32 over one staged 32-wide K tile.
// ---------------------------------------------------------------------------
__device__ __forceinline__ void mma_tile(const float* lA, const float* lB,
                                         int wave, int lm, int half, v8f acc[4]) {
  #pragma unroll
  for (int kk = 0; kk < KT; kk += 4) {
    // A frag: lane holds token row (wave*16+lm), K = kk + half*2 .. +1
    v2f a = *(const v2f*)(lA + (wave * 16 + lm) * KTP + kk + half * 2);
    #pragma unroll
    for (int t = 0; t < 4; ++t) {
      // B frag: lane holds expert col (t*16+lm), K = kk + half*2 .. +1
      v2f b = *(const v2f*)(lB + (t * 16 + lm) * KTP + kk + half * 2);
      acc[t] = __builtin_amdgcn_wmma_f32_16x16x4_f32(
          false, a, false, b, (short)0, acc[t], false, false);
    }
  }
}

#if ASYNC_LDS
// ---------------------------------------------------------------------------
// Issue one K-tile of async global->LDS b128 copies (8 per wave, 16 KB/block).
// ---------------------------------------------------------------------------
__device__ __forceinline__ void issue_tile(const float* __restrict__ X,
                                           const float* __restrict__ W,
                                           float* lA, float* lB,
                                           int row0, int k0, int tid) {
  #pragma unroll
  for (int i = 0; i < 4; ++i) {
    int idx = tid + i * THREADS;
    int r = idx >> 3, c4 = idx & 7;           // 8 float4 per row
    __builtin_amdgcn_global_load_async_to_lds_b128(
        GPTR(X + (size_t)(row0 + r) * H_DIM + k0 + c4 * 4),
        LPTR(lA + r * KTP + c4 * 4), 0, 0);
  }
  #pragma unroll
  for (int i = 0; i < 4; ++i) {
    int idx = tid + i * THREADS;
    int r = idx >> 3, c4 = idx & 7;
    __builtin_amdgcn_global_load_async_to_lds_b128(
        GPTR(W + (size_t)r * H_DIM + k0 + c4 * 4),
        LPTR(lB + r * KTP + c4 * 4), 0, 0);
  }
}
#endif

// ---------------------------------------------------------------------------
// Kernel 1: logits[T,E] = X[T,H] * W[E,H]^T  (f32 WMMA, deep async pipeline)
// ---------------------------------------------------------------------------
__global__ __launch_bounds__(THREADS)
void router_gemm_kernel(const float* __restrict__ X,
                        const float* __restrict__ W,
                        float* __restrict__ logits) {
  const int tid  = threadIdx.x;
  const int wave = tid >> 5;
  const int lane = tid & 31;
  const int half = lane >> 4;
  const int lm   = lane & 15;
  const int row0 = blockIdx.x * BM;

  v8f acc[4] = {};  // acc[t] covers experts t*16 .. t*16+15

#if ASYNC_LDS
  constexpr int ST = 6;                    // pipeline depth: 5 tiles in flight
  __shared__ float lA[ST][BM * KTP];       // 6 x 9216 B
  __shared__ float lB[ST][E_EXP * KTP];    // 6 x 9216 B   (total ~110 KB)

  #pragma unroll
  for (int s = 0; s < ST; ++s)
    issue_tile(X, W, lA[s], lB[s], row0, s * KT, tid);

  int b = 0;
  for (int t = 0; t < NT - ST; ++t) {
    WAIT_ASYNC(8 * (ST - 1));   // oldest stage complete (async done in order)
    __syncthreads();            // ... for all waves
    mma_tile(lA[b], lB[b], wave, lm, half, acc);
    __syncthreads();            // all waves done reading buffer b
    issue_tile(X, W, lA[b], lB[b], row0, (t + ST) * KT, tid);
    b = (b + 1 == ST) ? 0 : b + 1;
  }
  // Epilogue: drain last ST stages; explicit so wait immediates are constants.
  #define EPI(WCNT, LAST)                                       \
    {                                                           \
      WAIT_ASYNC(WCNT);                                         \
      __syncthreads();                                          \
      mma_tile(lA[b], lB[b], wave, lm, half, acc);              \
      if (!(LAST)) __syncthreads();                             \
      b = (b + 1 == ST) ? 0 : b + 1;                            \
    }
  EPI(40, 0) EPI(32, 0) EPI(24, 0) EPI(16, 0) EPI(8, 0) EPI(0, 1)
  #undef EPI
#else
  // Fallback: register-prefetch double buffer (loads overlap compute).
  __shared__ float lA[BM * KTP];
  __shared__ float lB[E_EXP * KTP];
  v4f ra[4], rb[4];
  #pragma unroll
  for (int i = 0; i < 4; ++i) {
    int idx = tid + i * THREADS, r = idx >> 3, c4 = idx & 7;
    ra[i] = *(const v4f*)(X + (size_t)(row0 + r) * H_DIM + c4 * 4);
    rb[i] = *(const v4f*)(W + (size_t)r * H_DIM + c4 * 4);
  }
  for (int t = 0; t < NT; ++t) {
    __syncthreads();  // previous compute done reading LDS
    #pragma unroll
    for (int i = 0; i < 4; ++i) {
      int idx = tid + i * THREADS, r = idx >> 3, c4 = idx & 7;
      *(v4f*)(&lA[r * KTP + c4 * 4]) = ra[i];
      *(v4f*)(&lB[r * KTP + c4 * 4]) = rb[i];
    }
    if (t + 1 < NT) {
      int k0 = (t + 1) * KT;  // issue next tile's loads; overlap with compute
      #pragma unroll
      for (int i = 0; i < 4; ++i) {
        int idx = tid + i * THREADS, r = idx >> 3, c4 = idx & 7;
        ra[i] = *(const v4f*)(X + (size_t)(row0 + r) * H_DIM + k0 + c4 * 4);
        rb[i] = *(const v4f*)(W + (size_t)r * H_DIM + k0 + c4 * 4);
      }
    }
    __syncthreads();
    mma_tile(lA, lB, wave, lm, half, acc);
  }
#endif

  // store C: VGPR r holds token row (half*8 + r), lane lm = expert col
  const int tokrow = row0 + wave * 16 + half * 8;
  #pragma unroll
  for (int r = 0; r < 8; ++r) {
    #pragma unroll
    for (int t = 0; t < 4; ++t) {
      logits[(size_t)(tokrow + r) * E_EXP + t * 16 + lm] = acc[t][r];
    }
  }
}

// ---------------------------------------------------------------------------
// Kernel 2: per-token top-2 + renormalized 2-way softmax.
// ---------------------------------------------------------------------------
__global__ __launch_bounds__(256)
void router_top2_kernel(const float* __restrict__ logits,
                        float* __restrict__ topw,
                        float* __restrict__ topi) {
  const int lane  = threadIdx.x & 31;
  const int warp  = threadIdx.x >> 5;
  const int token = blockIdx.x * 8 + warp;

  v2f v = *(const v2f*)(logits + (size_t)token * E_EXP + lane * 2);
  float m1, m2;
  int   i1, i2;
  if (v.x >= v.y) { m1 = v.x; i1 = lane * 2;     m2 = v.y; i2 = lane * 2 + 1; }
  else            { m1 = v.y; i1 = lane * 2 + 1; m2 = v.x; i2 = lane * 2;     }

  #pragma unroll
  for (int off = 16; off >= 1; off >>= 1) {
    float om1 = __shfl_xor(m1, off, 32);
    int   oi1 = __shfl_xor(i1, off, 32);
    float om2 = __shfl_xor(m2, off, 32);
    int   oi2 = __shfl_xor(i2, off, 32);
    bool  takeO = (om1 > m1) || (om1 == m1 && oi1 < i1);
    float w1 = takeO ? om1 : m1;  int w1i = takeO ? oi1 : i1;   // winner first
    float lf = takeO ? m1  : om1; int lfi = takeO ? i1  : oi1;  // loser first
    float ws = takeO ? om2 : m2;  int wsi = takeO ? oi2 : i2;   // winner second
    bool  secL = (lf > ws) || (lf == ws && lfi < wsi);
    m1 = w1;             i1 = w1i;
    m2 = secL ? lf : ws; i2 = secL ? lfi : wsi;
  }

  if (lane == 0) {
    float e     = __expf(m2 - m1);
    float denom = 1.0f + e;
    topw[token * 2 + 0] = 1.0f / denom;
    topw[token * 2 + 1] = e / denom;
    topi[token * 2 + 0] = (float)i1;
    topi[token * 2 + 1] = (float)i2;
  }
}

// ---------------------------------------------------------------------------
extern "C" void kernel_launch(void* const* d_in, const int* in_sizes, int n_in,
                              void* d_out, int out_size, void* d_ws, size_t ws_size,
                              hipStream_t stream) {
  const float* X = (const float*)d_in[0];   // hidden_states [16384, 4096]
  const float* W = (const float*)d_in[1];   // classifier W  [64, 4096]
  float* out = (float*)d_out;

  // d_out layout: topk_weight [T,2] | router_logits [T,E] | topk_index [T,2]
  float* topw   = out;
  float* logits = out + (size_t)T_TOK * 2;
  float* topi   = out + (size_t)T_TOK * 2 + (size_t)T_TOK * E_EXP;

  router_gemm_kernel<<<T_TOK / BM, THREADS, 0, stream>>>(X, W, logits);
  router_top2_kernel<<<T_TOK / 8, 256, 0, stream>>>(logits, topw, topi);
}